// GNNActorCritic_42855183680012
// MI455X (gfx1250) — compile-verified
//
#include <hip/hip_runtime.h>
#include <hip/hip_bf16.h>

// ---------------------------------------------------------------------------
// GNN actor-critic for MI455X (gfx1250).
// Dominant cost: streaming aW (20 x 4.19M f32 = 335 MB) -> HBM-bound ~15us,
// now read with b128 per-lane loads. x@W1 runs on the WMMA pipe with
// exact-precision V_WMMA_F32_16X16X4_F32.
// ---------------------------------------------------------------------------

#define GN 2048
#define GE 65536
#define GH1 32
#define GH2 20
#define GNN_SZ ((size_t)GN * (size_t)GN)   // 4194304

typedef __attribute__((ext_vector_type(2))) float v2f;
typedef __attribute__((ext_vector_type(8))) float v8f;

// ---------------- init: deg=1 (self loop), zero aggregation buffers --------
__global__ void gac_init_kernel(float* __restrict__ deg,
                                float* __restrict__ agg1,
                                float* __restrict__ agg2) {
  int idx = blockIdx.x * blockDim.x + threadIdx.x;
  if (idx < GN * GH1) agg1[idx] = 0.0f;
  if (idx < GN * GH2) agg2[idx] = 0.0f;
  if (idx < GN)       deg[idx]  = 1.0f;
}

// ---------------- degree accumulation over edges ---------------------------
__global__ void gac_deg_kernel(const int* __restrict__ ei, float* __restrict__ deg) {
  int e = blockIdx.x * blockDim.x + threadIdx.x;
  if (e >= GE) return;
  atomicAdd(&deg[ei[GE + e]], 1.0f);
}

__global__ void gac_dinv_kernel(const float* __restrict__ deg, float* __restrict__ dinv) {
  int i = blockIdx.x * blockDim.x + threadIdx.x;
  if (i >= GN) return;
  float d = deg[i];
  dinv[i] = (d > 0.0f) ? rsqrtf(d) : 0.0f;
}

// ---------------- GEMM1: h1pre = x @ W1  via f32 WMMA ----------------------
// One wave per 16-row tile; both 16-wide N tiles (H1=32) share the A fragment.
// f32 WMMA 16x16x4 layouts (wave32):
//   A 16x4 : lanes 0-15 rows M=lane, v0/v1 hold K=0/1; lanes 16-31 hold K=2/3
//   B 4x16 : v0 = (K=0, N=lane | K=2, N=lane-16), v1 = (K=1 | K=3)
//   C 16x16: lanes 0-15 cols N=lane rows 0..7 (v0..v7); lanes 16-31 rows 8..15
__global__ __launch_bounds__(32) void gac_gemm1_wmma(const float* __restrict__ x,
                                                     const float* __restrict__ W1,
                                                     float* __restrict__ h1pre) {
  const int lane = threadIdx.x;
  const int r0   = blockIdx.x * 16;
  const int lr   = lane & 15;
  const int kh   = lane >> 4;   // 0 or 1
#if __has_builtin(__builtin_amdgcn_wmma_f32_16x16x4_f32)
  v8f c0 = {}; v8f c1 = {};
  const float* xrow = x + (size_t)(r0 + lr) * GN;
  for (int k = 0; k < GN; k += 4) {
    const int ka = k + 2 * kh;
    v2f a = *(const v2f*)(xrow + ka);           // 8B load: x[row][ka..ka+1]
    v2f b0, b1;
    b0.x = W1[ka * GH1 + lr];        b0.y = W1[(ka + 1) * GH1 + lr];
    b1.x = W1[ka * GH1 + 16 + lr];   b1.y = W1[(ka + 1) * GH1 + 16 + lr];
    c0 = __builtin_amdgcn_wmma_f32_16x16x4_f32(false, a, false, b0, (short)0, c0, false, false);
    c1 = __builtin_amdgcn_wmma_f32_16x16x4_f32(false, a, false, b1, (short)0, c1, false, false);
  }
  for (int v = 0; v < 8; ++v) {
    const int row = r0 + v + 8 * kh;
    h1pre[row * GH1 + lr]      = c0[v];
    h1pre[row * GH1 + 16 + lr] = c1[v];
  }
#else
  // Scalar fallback (should not be taken on gfx1250 toolchains).
  const int col = lane;
  for (int m = 0; m < 16; ++m) {
    float acc = 0.0f;
    const float* xr = x + (size_t)(r0 + m) * GN;
    for (int k = 0; k < GN; ++k) acc += xr[k] * W1[k * GH1 + col];
    h1pre[(r0 + m) * GH1 + col] = acc;
  }
#endif
}

// ---------------- edge scatter-add: agg[dst] += h[src] * dinv[s]*dinv[d] ---
template <int F>
__global__ void gac_edge_agg_kernel(const int* __restrict__ ei,
                                    const float* __restrict__ dinv,
                                    const float* __restrict__ h,
                                    float* __restrict__ agg) {
  int idx = blockIdx.x * blockDim.x + threadIdx.x;
  if (idx >= GE * F) return;
  const int e = idx / F, k = idx % F;
  const int s = ei[e], d = ei[GE + e];
  const float w = dinv[s] * dinv[d];
  atomicAdd(&agg[d * F + k], h[s * F + k] * w);
}

// ---------------- self-loop + bias + relu ----------------------------------
template <int F>
__global__ void gac_relu_bias_kernel(const float* __restrict__ agg,
                                     const float* __restrict__ hpre,
                                     const float* __restrict__ dinv,
                                     const float* __restrict__ b,
                                     float* __restrict__ out) {
  int idx = blockIdx.x * blockDim.x + threadIdx.x;
  if (idx >= GN * F) return;
  const int i = idx / F, k = idx % F;
  const float v = agg[idx] + dinv[i] * dinv[i] * hpre[idx] + b[k];
  out[idx] = fmaxf(v, 0.0f);
}

// ---------------- GEMM2: h2pre = h1 @ W2 (2048x32 x 32x20, trivial) --------
__global__ void gac_gemm2_kernel(const float* __restrict__ h1,
                                 const float* __restrict__ W2,
                                 float* __restrict__ h2pre) {
  int idx = blockIdx.x * blockDim.x + threadIdx.x;
  if (idx >= GN * GH2) return;
  const int i = idx / GH2, j = idx % GH2;
  float acc = 0.0f;
  for (int k = 0; k < GH1; ++k) acc += h1[i * GH1 + k] * W2[k * GH2 + j];
  h2pre[idx] = acc;
}

// ---------------- column mean of h2 -> x_actor[20]; critic -> out[8] -------
__global__ __launch_bounds__(256) void gac_colmean_kernel(const float* __restrict__ h2,
                                                          const float* __restrict__ cW,
                                                          const float* __restrict__ cb,
                                                          float* __restrict__ xa,
                                                          float* __restrict__ out) {
  __shared__ float acc[GH2];
  const int tid = threadIdx.x;
  if (tid < GH2) acc[tid] = 0.0f;
  __syncthreads();
  float loc[GH2];
  for (int k = 0; k < GH2; ++k) loc[k] = 0.0f;
  for (int r = tid; r < GN; r += 256)
    for (int k = 0; k < GH2; ++k) loc[k] += h2[r * GH2 + k];
  for (int k = 0; k < GH2; ++k) atomicAdd(&acc[k], loc[k]);  // ds_add_f32
  __syncthreads();
  if (tid < GH2) xa[tid] = acc[tid] * (1.0f / (float)GN);
  if (tid == 0) {
    float c = cb[0];
    for (int k = 0; k < GH2; ++k) c += acc[k] * (1.0f / (float)GN) * cW[k];
    out[8] = c;
  }
}

// ---------------- per-row logits + softmax stats (streams aW, 335 MB) ------
// b128 per-lane loads: thread t owns columns {4t..4t+3} and {1024+4t..1024+4t+3},
// ascending within thread -> first-occurrence argmax semantics preserved.
__global__ __launch_bounds__(256) void gac_logits_kernel(const float* __restrict__ aW,
                                                         const float* __restrict__ ab,
                                                         const float* __restrict__ xa,
                                                         float* __restrict__ rowmax,
                                                         float* __restrict__ rowsum,
                                                         int* __restrict__ rowargmax) {
  __shared__ float ls[GN];
  __shared__ float xs[GH2];
  __shared__ float rv[256];
  __shared__ int   ri[256];
  const int tid = threadIdx.x;
  const size_t base = (size_t)blockIdx.x * GN;
  if (tid < GH2) xs[tid] = xa[tid];
  __syncthreads();
  for (int p = 0; p < 2; ++p) {
    const int j = tid * 4 + 1024 * p;
    float4 acc = *(const float4*)(ab + base + j);
    for (int k = 0; k < GH2; ++k) {
      const float* src = aW + (size_t)k * GNN_SZ + base + j;
      const float4 w = *(const float4*)src;          // global_load_b128
      acc.x += xs[k] * w.x;  acc.y += xs[k] * w.y;
      acc.z += xs[k] * w.z;  acc.w += xs[k] * w.w;
      __builtin_prefetch(src + 1024, 0, 1);          // global_prefetch_b8
    }
    *(float4*)(ls + j) = acc;
  }
  __syncthreads();
  // max + first-occurrence argmax
  float bv = -__builtin_inff(); int bi = 0;
  for (int p = 0; p < 2; ++p)
    for (int q = 0; q < 4; ++q) {
      const int j = tid * 4 + 1024 * p + q;
      const float v = ls[j];
      if (v > bv) { bv = v; bi = j; }
    }
  rv[tid] = bv; ri[tid] = bi;
  __syncthreads();
  for (int s = 128; s > 0; s >>= 1) {
    if (tid < s && (rv[tid + s] > rv[tid] ||
                    (rv[tid + s] == rv[tid] && ri[tid + s] < ri[tid]))) {
      rv[tid] = rv[tid + s]; ri[tid] = ri[tid + s];
    }
    __syncthreads();
  }
  __shared__ float ms; __shared__ int cs;
  if (tid == 0) { ms = rv[0]; cs = ri[0]; }
  __syncthreads();
  const float m = ms;
  float lsum = 0.0f;
  for (int p = 0; p < 2; ++p)
    for (int q = 0; q < 4; ++q) lsum += expf(ls[tid * 4 + 1024 * p + q] - m);
  rv[tid] = lsum;
  __syncthreads();
  for (int s = 128; s > 0; s >>= 1) {
    if (tid < s) rv[tid] += rv[tid + s];
    __syncthreads();
  }
  if (tid == 0) { rowmax[blockIdx.x] = m; rowsum[blockIdx.x] = rv[0]; rowargmax[blockIdx.x] = cs; }
}

// ---------------- final selection + log-probs ------------------------------
__global__ __launch_bounds__(256) void gac_finalize_kernel(const float* __restrict__ aW,
                                                           const float* __restrict__ ab,
                                                           const float* __restrict__ xa,
                                                           const float* __restrict__ rowmax,
                                                           const float* __restrict__ rowsum,
                                                           const int* __restrict__ rowargmax,
                                                           float* __restrict__ out) {
  __shared__ float rv[256];
  __shared__ int   ri[256];
  __shared__ float xs[GH2];
  __shared__ int   r1s, c1s, iAs, sel[4];
  __shared__ float vAs;
  const int tid = threadIdx.x;
  if (tid < GH2) xs[tid] = xa[tid];

  // Phase A: global best row (max prob = 1/rowsum, ties -> smaller row)
  float bv = -__builtin_inff(); int bi = 0;
  for (int r = tid; r < GN; r += 256) {
    const float p = 1.0f / rowsum[r];
    if (p > bv) { bv = p; bi = r; }
  }
  rv[tid] = bv; ri[tid] = bi;
  __syncthreads();
  for (int s = 128; s > 0; s >>= 1) {
    if (tid < s && (rv[tid + s] > rv[tid] ||
                    (rv[tid + s] == rv[tid] && ri[tid + s] < ri[tid]))) {
      rv[tid] = rv[tid + s]; ri[tid] = ri[tid + s];
    }
    __syncthreads();
  }
  if (tid == 0) { r1s = ri[0]; c1s = rowargmax[ri[0]]; }
  __syncthreads();
  const int r1 = r1s, c1 = c1s;
  const int idx1 = r1 * GN + c1;
  const size_t base1 = (size_t)r1 * GN;

  // Phase B: best among full rows strictly before r1
  bv = -__builtin_inff(); bi = 0;
  for (int r = tid; r < r1; r += 256) {
    const float p = 1.0f / rowsum[r];
    if (p > bv) { bv = p; bi = r; }
  }
  rv[tid] = bv; ri[tid] = bi;
  __syncthreads();
  for (int s = 128; s > 0; s >>= 1) {
    if (tid < s && (rv[tid + s] > rv[tid] ||
                    (rv[tid + s] == rv[tid] && ri[tid + s] < ri[tid]))) {
      rv[tid] = rv[tid + s]; ri[tid] = ri[tid + s];
    }
    __syncthreads();
  }
  if (tid == 0) { vAs = rv[0]; iAs = ri[0] * GN + rowargmax[ri[0]]; }
  __syncthreads();
  const float vA = vAs; const int iA = iAs;

  // Phase C: prefix of row r1 before c1 (recompute logits, 20-wide dots)
  const float m1 = rowmax[r1], is1 = 1.0f / rowsum[r1];
  bv = -__builtin_inff(); bi = 0;
  for (int j = tid; j < c1; j += 256) {
    float l = ab[base1 + j];
    for (int k = 0; k < GH2; ++k) l += xs[k] * aW[(size_t)k * GNN_SZ + base1 + j];
    const float p = expf(l - m1) * is1;
    if (p > bv) { bv = p; bi = j; }
  }
  rv[tid] = bv; ri[tid] = bi;
  __syncthreads();
  for (int s = 128; s > 0; s >>= 1) {
    if (tid < s && (rv[tid + s] > rv[tid] ||
                    (rv[tid + s] == rv[tid] && ri[tid + s] < ri[tid]))) {
      rv[tid] = rv[tid + s]; ri[tid] = ri[tid + s];
    }
    __syncthreads();
  }
  if (tid == 0) {
    const float vB = rv[0]; const int iB = r1 * GN + ri[0];
    int idx2;
    if (idx1 == 0)      idx2 = 0;            // masked all -inf -> argmax 0
    else if (vA >= vB)  idx2 = iA;           // tie -> smaller flat index (A)
    else                idx2 = iB;
    sel[0] = r1; sel[1] = c1; sel[2] = idx2 / GN; sel[3] = idx2 % GN;
  }
  __syncthreads();

  // Phase E: faithful bug -- index FLAT probs with (i,j) values => row 0 probs
  if (tid < 4) {
    const int t = sel[tid];
    float l = ab[t];
    for (int k = 0; k < GH2; ++k) l += xs[k] * aW[(size_t)k * GNN_SZ + t];
    const float p = expf(l - rowmax[0]) / rowsum[0];
    out[tid]     = (float)t;
    out[4 + tid] = -logf(p);
  }
}

// ---------------------------------------------------------------------------
extern "C" void kernel_launch(void* const* d_in, const int* in_sizes, int n_in,
                              void* d_out, int out_size, void* d_ws, size_t ws_size,
                              hipStream_t stream) {
  const float* x  = (const float*)d_in[0];
  const int*   ei = (const int*)d_in[1];
  const float* W1 = (const float*)d_in[2];
  const float* b1 = (const float*)d_in[3];
  const float* W2 = (const float*)d_in[4];
  const float* b2 = (const float*)d_in[5];
  const float* aW = (const float*)d_in[6];
  const float* ab = (const float*)d_in[7];
  const float* cW = (const float*)d_in[8];
  const float* cb = (const float*)d_in[9];
  float* out = (float*)d_out;

  float* ws     = (float*)d_ws;
  float* deg    = ws;                    // 2048
  float* dinv   = deg    + GN;           // 2048
  float* h1pre  = dinv   + GN;           // 65536
  float* agg1   = h1pre  + GN * GH1;     // 65536
  float* h1     = agg1   + GN * GH1;     // 65536
  float* h2pre  = h1     + GN * GH1;     // 40960
  float* agg2   = h2pre  + GN * GH2;     // 40960
  float* h2     = agg2   + GN * GH2;     // 40960
  float* xa     = h2     + GN * GH2;     // 32 (padded)
  float* rowmax = xa + 32;               // 2048
  float* rowsum = rowmax + GN;           // 2048
  int*   rowam  = (int*)(rowsum + GN);   // 2048

  gac_init_kernel<<<(GN * GH1 + 255) / 256, 256, 0, stream>>>(deg, agg1, agg2);
  gac_deg_kernel<<<(GE + 255) / 256, 256, 0, stream>>>(ei, deg);
  gac_dinv_kernel<<<(GN + 255) / 256, 256, 0, stream>>>(deg, dinv);
  gac_gemm1_wmma<<<GN / 16, 32, 0, stream>>>(x, W1, h1pre);
  gac_edge_agg_kernel<GH1><<<(GE * GH1 + 255) / 256, 256, 0, stream>>>(ei, dinv, h1pre, agg1);
  gac_relu_bias_kernel<GH1><<<(GN * GH1 + 255) / 256, 256, 0, stream>>>(agg1, h1pre, dinv, b1, h1);
  gac_gemm2_kernel<<<(GN * GH2 + 255) / 256, 256, 0, stream>>>(h1, W2, h2pre);
  gac_edge_agg_kernel<GH2><<<(GE * GH2 + 255) / 256, 256, 0, stream>>>(ei, dinv, h2pre, agg2);
  gac_relu_bias_kernel<GH2><<<(GN * GH2 + 255) / 256, 256, 0, stream>>>(agg2, h2pre, dinv, b2, h2);
  gac_colmean_kernel<<<1, 256, 0, stream>>>(h2, cW, cb, xa, out);
  gac_logits_kernel<<<GN, 256, 0, stream>>>(aW, ab, xa, rowmax, rowsum, rowam);
  gac_finalize_kernel<<<1, 256, 0, stream>>>(aW, ab, xa, rowmax, rowsum, rowam, out);
}